// Latent_36532991820082
// MI455X (gfx1250) — compile-verified
//
#include <hip/hip_runtime.h>
#include <math.h>
#include <stdint.h>

typedef __attribute__((ext_vector_type(2))) float v2f;
typedef __attribute__((ext_vector_type(8))) float v8f;

#define N_TOT 16384   // B*H*W
#define K_TOT 4096
#define D_TOT 256
#define HW    1024    // H*W
#define TM    128
#define TN    64
#define BK    16
#define LDP   20      // padded LDS row stride (floats): 16B-aligned rows,
                      // conflict-free b64 operand reads (20m mod 64 distinct)

// ---------------- small helper kernels ----------------

__global__ void init_cols_kernel(unsigned* colmax, unsigned* colmin, float* colsum) {
  int k = blockIdx.x * blockDim.x + threadIdx.x;
  if (k < K_TOT) { colmax[k] = 0u; colmin[k] = 0x7F800000u /*+inf*/; colsum[k] = 0.f; }
}

// zn[n] = sum_d z[b,d,h,w]^2 ; n = b*HW + hw.  Consecutive lanes -> consecutive hw,
// so each d-iteration is a fully coalesced 128B wave read.
__global__ void zn_kernel(const float* __restrict__ z, float* __restrict__ zn) {
  int n = blockIdx.x * blockDim.x + threadIdx.x;
  if (n >= N_TOT) return;
  int b = n >> 10, hw = n & 1023;
  const float* p = z + (size_t)b * D_TOT * HW + hw;
  float s = 0.f;
  #pragma unroll 4
  for (int d = 0; d < D_TOT; ++d) { float v = p[(size_t)d * HW]; s += v * v; }
  zn[n] = s;
}

// en[k] = sum_d e[k,d]^2 ; one wave per row, shuffle reduce (wave32).
__global__ void en_kernel(const float* __restrict__ e, float* __restrict__ en) {
  int row  = (blockIdx.x * blockDim.x + threadIdx.x) >> 5;
  int lane = threadIdx.x & 31;
  if (row >= K_TOT) return;
  const float* p = e + (size_t)row * D_TOT;
  float s = 0.f;
  for (int d = lane; d < D_TOT; d += 32) { float v = p[d]; s += v * v; }
  #pragma unroll
  for (int off = 16; off > 0; off >>= 1) s += __shfl_xor(s, off, 32);
  if (lane == 0) en[row] = s;
}

// nf[k] = max(0.999*norm_factor + 0.001*colmax, 1e-8)
// mt[k] = transform(colmin_dist, nf)  == column-min of transformed distance
// (transform is monotone-increasing in dist, so min commutes with it)
__global__ void nf_kernel(const float* __restrict__ norm_factor,
                          const unsigned* __restrict__ colmax,
                          const unsigned* __restrict__ colmin,
                          float* __restrict__ nf, float* __restrict__ mt,
                          const int* __restrict__ p_ptr) {
  int k = blockIdx.x * blockDim.x + threadIdx.x;
  if (k >= K_TOT) return;
  float p  = (float)p_ptr[0];
  float cm = __uint_as_float(colmax[k]);
  float f  = fmaxf(0.999f * norm_factor[k] + 0.001f * cm, 1e-8f);
  float dn = __uint_as_float(colmin[k]) / f;
  float t  = (dn < 1.f ? powf(dn, 2.f / p) : powf(dn, 2.f * p)) * f;
  nf[k] = f; mt[k] = t;
}

__global__ void loss_kernel(const float* __restrict__ mt,
                            const float* __restrict__ colsum,
                            float* __restrict__ out) {
  __shared__ float red[256];
  float s = 0.f;
  for (int k = threadIdx.x; k < K_TOT; k += 256) s += mt[k] - logf(colsum[k]);
  red[threadIdx.x] = s;
  __syncthreads();
  #pragma unroll
  for (int off = 128; off > 0; off >>= 1) {
    if (threadIdx.x < off) red[threadIdx.x] += red[threadIdx.x + off];
    __syncthreads();
  }
  if (threadIdx.x == 0) out[0] = red[0] / (float)K_TOT;
}

// LDS byte offset of a __shared__ object: flat LDS addresses map to the
// wave-relative LDS offset in their low 32 bits (CDNA5 aperture rule).
__device__ __forceinline__ unsigned lds_off(const void* p) {
  return (unsigned)(uintptr_t)p;
}

// ---------------- main GEMM pass (f32 WMMA, async-LDS double buffered) ----
// PASS 1: per-column max(belong*dist) and min(dist)
// PASS 2: per-column sum of exp(mt - transform(dist))
template <int PASS>
__global__ __launch_bounds__(256) void dist_pass_kernel(
    const float* __restrict__ z, const float* __restrict__ e,
    const float* __restrict__ belong,
    const float* __restrict__ zn, const float* __restrict__ en,
    unsigned* __restrict__ colmax, unsigned* __restrict__ colmin,
    const float* __restrict__ nf, const float* __restrict__ mt,
    float* __restrict__ colsum, const int* __restrict__ p_ptr) {
  __shared__ __align__(16) float As[2][TM][LDP];   // [buf][row][d-in-chunk]
  __shared__ __align__(16) float Bs[2][TN][LDP];   // [buf][code][d-in-chunk]
  __shared__ float redA[8][TN];
  __shared__ float redB[8][TN];

  const int tid  = threadIdx.x;
  const int lane = tid & 31;
  const int w    = tid >> 5;        // wave id 0..7 -> rows [16w, 16w+16)
  const int sel  = lane >> 4;       // lane half: picks K{0,1} vs K{2,3}
  const int l16  = lane & 15;
  const int n0   = blockIdx.x * TM;
  const int k0   = blockIdx.y * TN;

  // Async staging of one BK-chunk into buffer `buf` (no VGPR bounce, ASYNCcnt).
  auto stage = [&](int d0, int buf) {
    // A tile (128 x 16): for fixed d the 128 rows are contiguous hw addresses.
    #pragma unroll
    for (int i = 0; i < 8; ++i) {
      int idx = tid + 256 * i;
      int dd = idx >> 7, r = idx & 127;
      int n = n0 + r;
      int b = n >> 10, hw = n & 1023;
      const float* gp = z + (size_t)(b * D_TOT + d0 + dd) * HW + hw;
      unsigned lp = lds_off(&As[buf][r][dd]);
      asm volatile("global_load_async_to_lds_b32 %0, %1, off"
                   :: "v"(lp), "v"((unsigned long long)(uintptr_t)gp) : "memory");
    }
    // B tile (64 x 16): one 16B quad per thread, LDS rows are 16B aligned.
    {
      int c = tid >> 2, dd4 = (tid & 3) * 4;
      const float* gp = e + (size_t)(k0 + c) * D_TOT + d0 + dd4;
      unsigned lp = lds_off(&Bs[buf][c][dd4]);
      asm volatile("global_load_async_to_lds_b128 %0, %1, off"
                   :: "v"(lp), "v"((unsigned long long)(uintptr_t)gp) : "memory");
    }
  };

  v8f acc[4];
  #pragma unroll
  for (int t = 0; t < 4; ++t)
    #pragma unroll
    for (int i = 0; i < 8; ++i) acc[t][i] = 0.f;

  constexpr int NCHUNK = D_TOT / BK;
  stage(0, 0);
  asm volatile("s_wait_asynccnt 0x0" ::: "memory");
  __syncthreads();

  for (int c = 0; c < NCHUNK; ++c) {
    const int buf = c & 1;
    if (c + 1 < NCHUNK) stage((c + 1) * BK, buf ^ 1);  // overlap with compute

    // A 16x4 f32 layout: lane m=l16; VGPR0 holds K = kk+2*sel, VGPR1 = kk+2*sel+1
    // -> one b64 LDS read per operand. B (4x16) mirrored with n=l16.
    #pragma unroll
    for (int kk = 0; kk < BK; kk += 4) {
      v2f a = *(const v2f*)&As[buf][16 * w + l16][kk + 2 * sel];
      #pragma unroll
      for (int t = 0; t < 4; ++t) {
        v2f b = *(const v2f*)&Bs[buf][16 * t + l16][kk + 2 * sel];
        acc[t] = __builtin_amdgcn_wmma_f32_16x16x4_f32(
            false, a, false, b, (short)0, acc[t], false, false);
      }
    }

    if (c + 1 < NCHUNK) {
      asm volatile("s_wait_asynccnt 0x0" ::: "memory");
      __syncthreads();   // next buffer fully written, this buffer fully read
    }
  }

  // C/D layout: VGPR v, lane L -> row M = v + 8*sel, col N = l16
  float znv[8];
  #pragma unroll
  for (int v = 0; v < 8; ++v) znv[v] = zn[n0 + 16 * w + 8 * sel + v];

  if constexpr (PASS == 1) {
    float mxv[4], mnv[4];
    #pragma unroll
    for (int t = 0; t < 4; ++t) {
      int j = k0 + 16 * t + l16;
      float ej = en[j];
      const float* brow = belong + (size_t)(n0 + 16 * w + 8 * sel) * K_TOT + j;
      float mx = 0.f, mn = INFINITY;
      #pragma unroll
      for (int v = 0; v < 8; ++v) {
        float sq = znv[v] + ej - 2.f * acc[t][v];
        float d  = sqrtf(fmaxf(sq, 0.f));
        // belong is streamed exactly once (268 MB) -> non-temporal so it
        // doesn't evict the L2-resident z/e working set.
        float bl = __builtin_nontemporal_load(brow + (size_t)v * K_TOT);
        mx = fmaxf(mx, bl * d);
        mn = fminf(mn, d);
      }
      mx = fmaxf(mx, __shfl_xor(mx, 16, 32));
      mn = fminf(mn, __shfl_xor(mn, 16, 32));
      mxv[t] = mx; mnv[t] = mn;
    }
    if (lane < 16) {
      #pragma unroll
      for (int t = 0; t < 4; ++t) {
        redA[w][16 * t + l16] = mxv[t];
        redB[w][16 * t + l16] = mnv[t];
      }
    }
    __syncthreads();
    // one atomic per column per block (non-negative floats: uint order == float order)
    if (tid < TN) {
      float m = redA[0][tid];
      #pragma unroll
      for (int q = 1; q < 8; ++q) m = fmaxf(m, redA[q][tid]);
      atomicMax(&colmax[k0 + tid], __float_as_uint(m));
    } else if (tid < 2 * TN) {
      int c = tid - TN;
      float m = redB[0][c];
      #pragma unroll
      for (int q = 1; q < 8; ++q) m = fminf(m, redB[q][c]);
      atomicMin(&colmin[k0 + c], __float_as_uint(m));
    }
  } else {
    float p = (float)p_ptr[0];
    float e_lo = 2.f / p, e_hi = 2.f * p;
    float sv[4];
    #pragma unroll
    for (int t = 0; t < 4; ++t) {
      int j = k0 + 16 * t + l16;
      float ej = en[j], nfj = nf[j], mtj = mt[j];
      float s = 0.f;
      #pragma unroll
      for (int v = 0; v < 8; ++v) {
        float sq = znv[v] + ej - 2.f * acc[t][v];
        float d  = sqrtf(fmaxf(sq, 0.f));
        float dn = d / nfj;
        float tv = (dn < 1.f ? powf(dn, e_lo) : powf(dn, e_hi)) * nfj;
        s += expf(mtj - tv);   // mtj <= tv always -> no overflow
      }
      s += __shfl_xor(s, 16, 32);
      sv[t] = s;
    }
    if (lane < 16) {
      #pragma unroll
      for (int t = 0; t < 4; ++t) redA[w][16 * t + l16] = sv[t];
    }
    __syncthreads();
    if (tid < TN) {
      float s = redA[0][tid];
      #pragma unroll
      for (int q = 1; q < 8; ++q) s += redA[q][tid];
      atomicAdd(&colsum[k0 + tid], s);
    }
  }
}

// ---------------- host launcher ----------------

extern "C" void kernel_launch(void* const* d_in, const int* in_sizes, int n_in,
                              void* d_out, int out_size, void* d_ws, size_t ws_size,
                              hipStream_t stream) {
  (void)in_sizes; (void)n_in; (void)out_size; (void)ws_size;
  const float* z           = (const float*)d_in[0];
  const float* e           = (const float*)d_in[1];
  const float* belong      = (const float*)d_in[2];
  const float* norm_factor = (const float*)d_in[3];
  const int*   p_ptr       = (const int*)d_in[4];

  float*    ws     = (float*)d_ws;              // ~160 KB of scratch total
  float*    zn     = ws;                        // N_TOT
  float*    en     = zn + N_TOT;                // K_TOT
  unsigned* colmax = (unsigned*)(en + K_TOT);   // K_TOT
  unsigned* colmin = colmax + K_TOT;            // K_TOT
  float*    nf     = (float*)(colmin + K_TOT);  // K_TOT
  float*    mt     = nf + K_TOT;                // K_TOT
  float*    colsum = mt + K_TOT;                // K_TOT

  init_cols_kernel<<<(K_TOT + 255) / 256, 256, 0, stream>>>(colmax, colmin, colsum);
  zn_kernel<<<N_TOT / 256, 256, 0, stream>>>(z, zn);
  en_kernel<<<K_TOT / 8, 256, 0, stream>>>(e, en);

  dim3 grid(N_TOT / TM, K_TOT / TN);
  dist_pass_kernel<1><<<grid, 256, 0, stream>>>(z, e, belong, zn, en,
                                                colmax, colmin,
                                                nullptr, nullptr, nullptr, p_ptr);
  nf_kernel<<<(K_TOT + 255) / 256, 256, 0, stream>>>(norm_factor, colmax, colmin,
                                                     nf, mt, p_ptr);
  dist_pass_kernel<2><<<grid, 256, 0, stream>>>(z, e, nullptr, zn, en,
                                                nullptr, nullptr,
                                                nf, mt, colsum, p_ptr);
  loss_kernel<<<1, 256, 0, stream>>>(mt, colsum, (float*)d_out);
}